// EmbedMeanField_36051955483067
// MI455X (gfx1250) — compile-verified
//
#include <hip/hip_runtime.h>

typedef __attribute__((ext_vector_type(2))) float v2f;
typedef __attribute__((ext_vector_type(4))) float v4f;
typedef __attribute__((ext_vector_type(8))) float v8f;

#define LAT 128

__device__ __forceinline__ float leaky(float x) { return x > 0.0f ? x : 0.01f * x; }

__device__ __forceinline__ void atomic_add_f32(float* p, float v) {
    __hip_atomic_fetch_add(p, v, __ATOMIC_RELAXED, __HIP_MEMORY_SCOPE_AGENT);
}

// Y_tile[16, 128] += X[m0:m0+16, 0:K] @ W[128, K]^T  via V_WMMA_F32_16X16X4_F32.
// A 16x4: lane<16 -> M=lane, K={0,1}; lane>=16 -> M=lane-16, K={2,3}. B 4x16 symmetric.
// C/D: 8 VGPRs, M=r (lanes 0-15) / M=r+8 (lanes 16-31).
// B fragments batched into b[8]: compiler emits an s_clause of 8 loads, then 8 WMMAs
// drain with staggered s_wait_loadcnt (verified in disasm).
template <int K>
__device__ __forceinline__ void wmma_tile(const float* __restrict__ X,
                                          const float* __restrict__ W,
                                          int m0, v8f c[8]) {
    const int lane = threadIdx.x & 31;
    const int r    = lane & 15;
    const int kb   = (lane >> 4) << 1;          // 0 or 2
    const float* xrow = X + (size_t)(m0 + r) * K;
    const float* wrow = W + (size_t)r * K;
    for (int k = 0; k < K; k += 4) {
        v2f a = *(const v2f*)(xrow + k + kb);   // 8B aligned: k+kb even
        v2f b[8];
#pragma unroll
        for (int nt = 0; nt < 8; ++nt)
            b[nt] = *(const v2f*)(wrow + (size_t)nt * 16 * K + k + kb);
#pragma unroll
        for (int nt = 0; nt < 8; ++nt)
            c[nt] = __builtin_amdgcn_wmma_f32_16x16x4_f32(
                false, a, false, b[nt], (short)0, c[nt], false, false);
    }
}

__device__ __forceinline__ void init_bias(const float* __restrict__ bias, v8f c[8]) {
    const int lane = threadIdx.x & 31;
    const int ccol = lane & 15;
#pragma unroll
    for (int nt = 0; nt < 8; ++nt) {
        float bv = bias[nt * 16 + ccol];
#pragma unroll
        for (int rr = 0; rr < 8; ++rr) c[nt][rr] = bv;
    }
}

// msg = node_feats @ Wn^T + bn
__global__ void k_node_linear(const float* __restrict__ X, const float* __restrict__ W,
                              const float* __restrict__ bias, float* __restrict__ Y,
                              int nTiles) {
    int tile = blockIdx.x * (blockDim.x >> 5) + (threadIdx.x >> 5);
    if (tile >= nTiles) return;
    const int lane = threadIdx.x & 31;
    const int ccol = lane & 15;
    const int crow = (lane >> 4) << 3;
    v8f c[8];
    init_bias(bias, c);
    const int m0 = tile * 16;
    wmma_tile<LAT>(X, W, m0, c);
#pragma unroll
    for (int nt = 0; nt < 8; ++nt)
#pragma unroll
        for (int rr = 0; rr < 8; ++rr)
            Y[(size_t)(m0 + crow + rr) * LAT + nt * 16 + ccol] = c[nt][rr];
}

// msg[e2n_dst[e]] += edge_feats[e] @ We^T + be   (one-time atomic scatter)
__global__ void k_edge_embed_scatter(const float* __restrict__ EF, const float* __restrict__ W,
                                     const float* __restrict__ bias, const int* __restrict__ e2n,
                                     float* __restrict__ msg, int eTiles) {
    int tile = blockIdx.x * (blockDim.x >> 5) + (threadIdx.x >> 5);
    if (tile >= eTiles) return;
    const int lane = threadIdx.x & 31;
    const int ccol = lane & 15;
    const int crow = (lane >> 4) << 3;
    v8f c[8];
    init_bias(bias, c);
    const int e0 = tile * 16;
    wmma_tile<32>(EF, W, e0, c);
#pragma unroll
    for (int rr = 0; rr < 8; ++rr) {
        int d = e2n[e0 + crow + rr];
        float* row = msg + (size_t)d * LAT;
#pragma unroll
        for (int nt = 0; nt < 8; ++nt)
            atomic_add_f32(row + nt * 16 + ccol, c[nt][rr]);
    }
}

__global__ void k_leaky_copy(const float* __restrict__ in, float* __restrict__ out, int n4) {
    int i = blockIdx.x * blockDim.x + threadIdx.x;
    if (i >= n4) return;
    v4f v = ((const v4f*)in)[i];
    v4f r;
    r.x = leaky(v.x); r.y = leaky(v.y); r.z = leaky(v.z); r.w = leaky(v.w);
    ((v4f*)out)[i] = r;
}

__global__ void k_zero_f32(float* __restrict__ p, int n4) {
    int i = blockIdx.x * blockDim.x + threadIdx.x;
    if (i < n4) ((v4f*)p)[i] = (v4f){0.0f, 0.0f, 0.0f, 0.0f};
}

// ---------------- CSR build (sort edge sources by destination, once per launch) ----------------
__global__ void k_zero_i32(int* __restrict__ p, int n) {
    int i = blockIdx.x * blockDim.x + threadIdx.x;
    if (i < n) p[i] = 0;
}

__global__ void k_count(const int* __restrict__ dst, int* __restrict__ counts, int E) {
    int e = blockIdx.x * blockDim.x + threadIdx.x;
    if (e < E) atomicAdd(&counts[dst[e]], 1);
}

// single-block exclusive scan: counts[N] -> row_ptr[N+1] (chunked + LDS Hillis-Steele)
__global__ void __launch_bounds__(1024) k_scan(const int* __restrict__ counts,
                                               int* __restrict__ row_ptr, int N) {
    __shared__ int partial[1024];
    const int tid = threadIdx.x;
    const int chunk = (N + 1023) / 1024;
    const int base = tid * chunk;
    int sum = 0;
    for (int i = 0; i < chunk; ++i)
        if (base + i < N) sum += counts[base + i];
    partial[tid] = sum;
    __syncthreads();
    for (int off = 1; off < 1024; off <<= 1) {
        int v = (tid >= off) ? partial[tid - off] : 0;
        __syncthreads();
        partial[tid] += v;
        __syncthreads();
    }
    int run = partial[tid] - sum;   // exclusive prefix of this chunk
    for (int i = 0; i < chunk; ++i) {
        if (base + i < N) {
            row_ptr[base + i] = run;
            run += counts[base + i];
        }
    }
    if (tid == 1023) row_ptr[N] = partial[1023];
}

__global__ void k_copy_i32(const int* __restrict__ in, int* __restrict__ out, int n) {
    int i = blockIdx.x * blockDim.x + threadIdx.x;
    if (i < n) out[i] = in[i];
}

// Store the permuted SOURCE node id directly (not the edge id): the gather loop
// then needs a single indirection instead of eids[i] -> src[eids[i]].
__global__ void k_fill_src(const int* __restrict__ dst, const int* __restrict__ src,
                           int* __restrict__ cursor, int* __restrict__ srcs, int E) {
    int e = blockIdx.x * blockDim.x + threadIdx.x;
    if (e < E) {
        int p = atomicAdd(&cursor[dst[e]], 1);
        srcs[p] = src[e];
    }
}

// pool[n] = sum over incoming edges of cur[src]  — pure gather, no fp atomics.
// One wave per node; degree loop is wave-uniform; lane covers 4 columns.
__global__ void k_gather_csr(const float* __restrict__ cur, const int* __restrict__ row_ptr,
                             const int* __restrict__ srcs, float* __restrict__ pool, int N) {
    int n = blockIdx.x * (blockDim.x >> 5) + (threadIdx.x >> 5);
    if (n >= N) return;
    const int lane = threadIdx.x & 31;
    const int beg = row_ptr[n];
    const int end = row_ptr[n + 1];
    v4f acc = (v4f){0.0f, 0.0f, 0.0f, 0.0f};
    for (int i = beg; i < end; ++i) {
        int s = srcs[i];
        acc += *(const v4f*)(cur + (size_t)s * LAT + lane * 4);
    }
    *(v4f*)(pool + (size_t)n * LAT + lane * 4) = acc;
}

// cur = leaky(pool @ Wc^T + bc + msg)
__global__ void k_conv(const float* __restrict__ pool, const float* __restrict__ W,
                       const float* __restrict__ bias, const float* __restrict__ msg,
                       float* __restrict__ cur, int nTiles) {
    int tile = blockIdx.x * (blockDim.x >> 5) + (threadIdx.x >> 5);
    if (tile >= nTiles) return;
    const int lane = threadIdx.x & 31;
    const int ccol = lane & 15;
    const int crow = (lane >> 4) << 3;
    v8f c[8];
    init_bias(bias, c);
    const int m0 = tile * 16;
    wmma_tile<LAT>(pool, W, m0, c);
#pragma unroll
    for (int nt = 0; nt < 8; ++nt)
#pragma unroll
        for (int rr = 0; rr < 8; ++rr) {
            size_t idx = (size_t)(m0 + crow + rr) * LAT + nt * 16 + ccol;
            cur[idx] = leaky(c[nt][rr] + msg[idx]);
        }
}

// out[graph_ids[node]] += leaky(cur @ Wo^T + bo)
__global__ void k_out_scatter(const float* __restrict__ cur, const float* __restrict__ W,
                              const float* __restrict__ bias, const int* __restrict__ gid,
                              float* __restrict__ out, int nTiles) {
    int tile = blockIdx.x * (blockDim.x >> 5) + (threadIdx.x >> 5);
    if (tile >= nTiles) return;
    const int lane = threadIdx.x & 31;
    const int ccol = lane & 15;
    const int crow = (lane >> 4) << 3;
    v8f c[8];
    init_bias(bias, c);
    const int m0 = tile * 16;
    wmma_tile<LAT>(cur, W, m0, c);
#pragma unroll
    for (int rr = 0; rr < 8; ++rr) {
        int g = gid[m0 + crow + rr];
        float* row = out + (size_t)g * LAT;
#pragma unroll
        for (int nt = 0; nt < 8; ++nt)
            atomic_add_f32(row + nt * 16 + ccol, leaky(c[nt][rr]));
    }
}

__global__ void k_leaky_inplace(float* __restrict__ p, int n) {
    int i = blockIdx.x * blockDim.x + threadIdx.x;
    if (i < n) p[i] = leaky(p[i]);
}

extern "C" void kernel_launch(void* const* d_in, const int* in_sizes, int n_in,
                              void* d_out, int out_size, void* d_ws, size_t ws_size,
                              hipStream_t stream) {
    const float* node_feats = (const float*)d_in[0];
    const float* edge_feats = (const float*)d_in[1];
    const int*   edge_src   = (const int*)d_in[2];
    const int*   edge_dst   = (const int*)d_in[3];
    const int*   e2n_dst    = (const int*)d_in[4];
    const int*   graph_ids  = (const int*)d_in[5];
    const float* wn_W = (const float*)d_in[6];
    const float* wn_b = (const float*)d_in[7];
    const float* we_W = (const float*)d_in[8];
    const float* we_b = (const float*)d_in[9];
    const float* wc_W = (const float*)d_in[10];
    const float* wc_b = (const float*)d_in[11];
    const float* wo_W = (const float*)d_in[12];
    const float* wo_b = (const float*)d_in[13];
    float* out = (float*)d_out;

    const int N = in_sizes[0] / LAT;   // 100000
    const int E = in_sizes[2];         // 1600000

    // workspace layout
    size_t fElems = (size_t)N * LAT;
    size_t need = 3 * fElems * sizeof(float)                      // msg, cur, pool
                + (size_t)N * sizeof(int)                         // counts/cursor
                + (size_t)(N + 1) * sizeof(int)                   // row_ptr
                + (size_t)E * sizeof(int) + 64;                   // srcs (dst-sorted src ids)
    if (ws_size < need) return;

    float* msg  = (float*)d_ws;
    float* cur  = msg + fElems;
    float* pool = cur + fElems;
    int* counts  = (int*)(pool + fElems);
    int* row_ptr = counts + N;
    int* srcs    = row_ptr + (N + 1);

    const int nTiles = N / 16;
    const int eTiles = E / 16;
    const dim3 blk(256);                       // 8 waves / block
    const int wpb = 8;
    const int gNodeTiles = (nTiles + wpb - 1) / wpb;
    const int gEdgeTiles = (eTiles + wpb - 1) / wpb;
    const int nElem4 = N * LAT / 4;
    const int gElem4 = (nElem4 + 255) / 256;
    const int gE = (E + 255) / 256;
    const int gN = (N + 255) / 256;
    const int gNodeWave = (N + wpb - 1) / wpb;

    // ---- build dst-sorted CSR of source ids (reused by all 3 levels) ----
    k_zero_i32<<<gN, blk, 0, stream>>>(counts, N);
    k_count<<<gE, blk, 0, stream>>>(edge_dst, counts, E);
    k_scan<<<1, 1024, 0, stream>>>(counts, row_ptr, N);
    k_copy_i32<<<gN, blk, 0, stream>>>(row_ptr, counts, N);   // counts becomes fill cursor
    k_fill_src<<<gE, blk, 0, stream>>>(edge_dst, edge_src, counts, srcs, E);

    // ---- input_message = node linear + scattered edge linear ----
    k_node_linear<<<gNodeTiles, blk, 0, stream>>>(node_feats, wn_W, wn_b, msg, nTiles);
    k_edge_embed_scatter<<<gEdgeTiles, blk, 0, stream>>>(edge_feats, we_W, we_b, e2n_dst, msg, eTiles);
    k_leaky_copy<<<gElem4, blk, 0, stream>>>(msg, cur, nElem4);

    // ---- 3 message-passing levels: CSR gather + WMMA conv ----
    for (int lv = 0; lv < 3; ++lv) {
        k_gather_csr<<<gNodeWave, blk, 0, stream>>>(cur, row_ptr, srcs, pool, N);
        k_conv<<<gNodeTiles, blk, 0, stream>>>(pool, wc_W, wc_b, msg, cur, nTiles);
    }

    // ---- readout ----
    k_zero_f32<<<(out_size / 4 + 255) / 256, blk, 0, stream>>>(out, out_size / 4);
    k_out_scatter<<<gNodeTiles, blk, 0, stream>>>(cur, wo_W, wo_b, graph_ids, out, nTiles);
    k_leaky_inplace<<<(out_size + 255) / 256, blk, 0, stream>>>(out, out_size);
}